// Model_88656714925581
// MI455X (gfx1250) — compile-verified
//
#include <hip/hip_runtime.h>
#include <math.h>

typedef __attribute__((ext_vector_type(2))) float v2f;
typedef __attribute__((ext_vector_type(8))) float v8f;

#define B_SZ 2
#define CK   64
#define NM   32768
#define NQ   4096
#define CV   256
#define TOPK 20
#define QT   32            // queries per workgroup (two WMMA N-tiles)
#define NWAVE 8
#define MT   (NWAVE * 16)  // memory rows per workgroup iteration
#define CAP  144           // candidate buffer per query (>= per-iter max of 128)

// ---------------------------------------------------------------------------
// Kernel 1: column norms  a[b][m] = sum_c mk[b,c,m]^2
// ---------------------------------------------------------------------------
__global__ __launch_bounds__(256)
void norms_kernel(const float* __restrict__ mk, float* __restrict__ norms) {
    int i = blockIdx.x * blockDim.x + threadIdx.x;   // [0, B*NM)
    if (i >= B_SZ * NM) return;
    int b = i / NM, m = i % NM;
    const float* p = mk + (size_t)b * CK * NM + m;
    float s = 0.f;
#pragma unroll 8
    for (int c = 0; c < CK; ++c) {
        float v = p[(size_t)c * NM];
        s += v * v;
    }
    norms[i] = s;
}

// ---------------------------------------------------------------------------
// Kernel 2: WMMA affinity GEMM + streaming top-20 + softmax
//   aff[m,q] = (2*dot - ||mk||^2)/8 = 0.25*dot - 0.125*norm
// qk fragments pre-scaled by 0.25; the -0.125*norm term is folded in as an
// extra rank-1 WMMA (A_extra = [norm,0,0,0], B_extra = [-0.125,0,0,0]), so
// the accumulator holds the final affinity directly.
// One workgroup: 32 queries (2 N-subtiles), 8 waves x 16 memory rows/iter.
// ---------------------------------------------------------------------------
__global__ __launch_bounds__(256)
void topk_kernel(const float* __restrict__ mk, const float* __restrict__ qk,
                 const float* __restrict__ norms,
                 float* __restrict__ w_out, int* __restrict__ i_out) {
    __shared__ float s_tval[QT][TOPK];
    __shared__ int   s_tidx[QT][TOPK];
    __shared__ float s_thr[QT];
    __shared__ int   s_cnt[QT];
    __shared__ float s_cval[QT][CAP];
    __shared__ int   s_cidx[QT][CAP];

    const int tid  = threadIdx.x;
    const int lane = tid & 31;
    const int wave = tid >> 5;
    const int b    = blockIdx.y;
    const int q0   = blockIdx.x * QT;

    if (tid < QT) {
        s_thr[tid] = -INFINITY;
        s_cnt[tid] = 0;
        for (int j = 0; j < TOPK; ++j) { s_tval[tid][j] = -INFINITY; s_tidx[tid][j] = 0; }
    }
    __syncthreads();

    // ISA 32-bit A(16x4) operand layout: lanes 0-15 -> M rows, VGPR v holds
    // K = v + 2*(lane>=16). B(4x16) mirrors with lanes 0-15 -> N cols.
    const int ml = lane & 15;            // M-row (A) / N-col (B) within tile
    const int kh = (lane >> 4) * 2;      // K offset for upper half-wave

    // qk (B-matrix) fragments for both query sub-tiles, pre-scaled by 0.25.
    v2f Bf[2][16];
#pragma unroll
    for (int s = 0; s < 2; ++s) {
        const float* qb = qk + (size_t)b * CK * NQ + q0 + s * 16 + ml;
#pragma unroll
        for (int j = 0; j < 16; ++j) {
            Bf[s][j].x = 0.25f * qb[(size_t)(j * 4 + kh + 0) * NQ];
            Bf[s][j].y = 0.25f * qb[(size_t)(j * 4 + kh + 1) * NQ];
        }
    }
    // Norm-fold B fragment: B_extra[k=0][*] = -0.125, rest 0.
    v2f Bn;
    Bn.x = (lane < 16) ? -0.125f : 0.0f;
    Bn.y = 0.0f;

    const float* mb = mk + (size_t)b * CK * NM;
    const float* nb = norms + (size_t)b * NM;

    for (int it = 0; it < NM / MT; ++it) {
        const int m0 = it * MT + wave * 16;

        // A-matrix fragments for this wave's 16 memory rows.
        const float* ap = mb + m0 + ml;
        v2f Af[16];
#pragma unroll
        for (int j = 0; j < 16; ++j) {
            Af[j].x = ap[(size_t)(j * 4 + kh + 0) * NM];
            Af[j].y = ap[(size_t)(j * 4 + kh + 1) * NM];
        }
        // Norm-fold A fragment: A_extra[m][k=0] = norm[m], rest 0.
        v2f An;
        float nv = nb[m0 + ml];
        An.x = (lane < 16) ? nv : 0.0f;
        An.y = 0.0f;

        v8f C0 = {}, C1 = {};
#pragma unroll
        for (int j = 0; j < 16; ++j) {
            C0 = __builtin_amdgcn_wmma_f32_16x16x4_f32(
                     false, Af[j], false, Bf[0][j], (short)0, C0, false, false);
            C1 = __builtin_amdgcn_wmma_f32_16x16x4_f32(
                     false, Af[j], false, Bf[1][j], (short)0, C1, false, false);
        }
        C0 = __builtin_amdgcn_wmma_f32_16x16x4_f32(
                 false, An, false, Bn, (short)0, C0, false, false);
        C1 = __builtin_amdgcn_wmma_f32_16x16x4_f32(
                 false, An, false, Bn, (short)0, C1, false, false);

        // C/D layout: VGPR r, lane L -> M = r + 8*(L>=16), N = L & 15.
        const int mrow = m0 + ((lane >> 4) << 3);
#pragma unroll
        for (int s = 0; s < 2; ++s) {
            const int   ql  = s * 16 + ml;
            const float thr = s_thr[ql];
#pragma unroll
            for (int r = 0; r < 8; ++r) {
                float val = (s == 0) ? C0[r] : C1[r];
                if (val > thr) {
                    int pos = atomicAdd(&s_cnt[ql], 1);
                    if (pos < CAP) { s_cval[ql][pos] = val; s_cidx[ql][pos] = mrow + r; }
                }
            }
        }
        __syncthreads();

        // Merge candidates into per-query top-20, raise threshold.
        if (tid < QT) {
            int n = s_cnt[tid]; if (n > CAP) n = CAP;
            if (n > 0) {
                float mn = s_tval[tid][0]; int mp = 0;
                for (int j = 1; j < TOPK; ++j)
                    if (s_tval[tid][j] < mn) { mn = s_tval[tid][j]; mp = j; }
                for (int i = 0; i < n; ++i) {
                    float v = s_cval[tid][i];
                    if (v > mn) {
                        s_tval[tid][mp] = v;
                        s_tidx[tid][mp] = s_cidx[tid][i];
                        mn = s_tval[tid][0]; mp = 0;
                        for (int j = 1; j < TOPK; ++j)
                            if (s_tval[tid][j] < mn) { mn = s_tval[tid][j]; mp = j; }
                    }
                }
                s_thr[tid] = mn;
                s_cnt[tid] = 0;
            }
        }
        __syncthreads();
    }

    // Softmax over the 20 survivors; emit (weight, index) pairs.
    if (tid < QT) {
        float mx = -INFINITY;
        for (int j = 0; j < TOPK; ++j) mx = fmaxf(mx, s_tval[tid][j]);
        float e[TOPK], sum = 0.f;
        for (int j = 0; j < TOPK; ++j) { e[j] = __expf(s_tval[tid][j] - mx); sum += e[j]; }
        float inv = 1.f / sum;
        size_t base = ((size_t)b * NQ + q0 + tid) * TOPK;
        for (int j = 0; j < TOPK; ++j) {
            w_out[base + j] = e[j] * inv;
            i_out[base + j] = s_tidx[tid][j];
        }
    }
}

// ---------------------------------------------------------------------------
// Kernel 3a: tiled transpose mv[b][c][m] -> mvT[b][m][c] (gather coalescing)
// ---------------------------------------------------------------------------
__global__ __launch_bounds__(256)
void transpose_kernel(const float* __restrict__ in, float* __restrict__ outT) {
    __shared__ float t[32][33];
    const int b  = blockIdx.z;
    const int m0 = blockIdx.x * 32;
    const int c0 = blockIdx.y * 32;
    const int tx = threadIdx.x & 31;
    const int ty = threadIdx.x >> 5;      // 8 rows per pass
    const float* ip = in  + (size_t)b * CV * NM;
    float*       op = outT + (size_t)b * NM * CV;
#pragma unroll
    for (int j = 0; j < 32; j += 8)
        t[ty + j][tx] = ip[(size_t)(c0 + ty + j) * NM + m0 + tx];
    __syncthreads();
#pragma unroll
    for (int j = 0; j < 32; j += 8)
        op[(size_t)(m0 + ty + j) * CV + c0 + tx] = t[tx][ty + j];
}

// ---------------------------------------------------------------------------
// Kernel 3b: coalesced sparse readout from transposed mv
// ---------------------------------------------------------------------------
__global__ __launch_bounds__(CV)
void readout_t_kernel(const float* __restrict__ t0, const float* __restrict__ t1,
                      const float* __restrict__ w_in, const int* __restrict__ i_in,
                      float* __restrict__ out) {
    __shared__ float sw[TOPK];
    __shared__ int   si[TOPK];
    const int bq = blockIdx.x;
    const int b  = bq / NQ, q = bq % NQ;
    const int c  = threadIdx.x;
    if (c < TOPK) {
        sw[c] = w_in[(size_t)bq * TOPK + c];
        si[c] = i_in[(size_t)bq * TOPK + c];
    }
    __syncthreads();
    const float* p0 = t0 + (size_t)b * NM * CV;
    const float* p1 = t1 + (size_t)b * NM * CV;
    float a0 = 0.f, a1 = 0.f;
#pragma unroll
    for (int j = 0; j < TOPK; ++j) {
        size_t base = (size_t)si[j] * CV + c;   // contiguous across threads
        a0 += sw[j] * p0[base];
        a1 += sw[j] * p1[base];
    }
    size_t o = ((size_t)b * CV + c) * NQ + q;
    out[o] = a0;
    out[(size_t)B_SZ * CV * NQ + o] = a1;
}

// ---------------------------------------------------------------------------
// Kernel 3 fallback: direct (strided) sparse readout
// ---------------------------------------------------------------------------
__global__ __launch_bounds__(CV)
void readout_kernel(const float* __restrict__ mv0, const float* __restrict__ mv1,
                    const float* __restrict__ w_in, const int* __restrict__ i_in,
                    float* __restrict__ out) {
    __shared__ float sw[TOPK];
    __shared__ int   si[TOPK];
    const int bq = blockIdx.x;
    const int b  = bq / NQ, q = bq % NQ;
    const int c  = threadIdx.x;
    if (c < TOPK) {
        sw[c] = w_in[(size_t)bq * TOPK + c];
        si[c] = i_in[(size_t)bq * TOPK + c];
    }
    __syncthreads();
    const float* p0 = mv0 + ((size_t)b * CV + c) * NM;
    const float* p1 = mv1 + ((size_t)b * CV + c) * NM;
    float a0 = 0.f, a1 = 0.f;
#pragma unroll
    for (int j = 0; j < TOPK; ++j) {
        float w = sw[j]; int m = si[j];
        a0 += w * p0[m];
        a1 += w * p1[m];
    }
    size_t o = ((size_t)b * CV + c) * NQ + q;
    out[o] = a0;
    out[(size_t)B_SZ * CV * NQ + o] = a1;
}

// ---------------------------------------------------------------------------
extern "C" void kernel_launch(void* const* d_in, const int* in_sizes, int n_in,
                              void* d_out, int out_size, void* d_ws, size_t ws_size,
                              hipStream_t stream) {
    const float* mk  = (const float*)d_in[0];
    const float* qk  = (const float*)d_in[1];
    const float* mv0 = (const float*)d_in[2];
    const float* mv1 = (const float*)d_in[3];
    float* out = (float*)d_out;

    // Workspace layout: norms | weights | idx | (optional) mv0T | mv1T
    float* norms = (float*)d_ws;
    float* w_ws  = norms + (size_t)B_SZ * NM;
    int*   i_ws  = (int*)(w_ws + (size_t)B_SZ * NQ * TOPK);
    const size_t base_bytes = ((size_t)B_SZ * NM + 2u * (size_t)B_SZ * NQ * TOPK) * 4u;
    const size_t t_elems    = (size_t)B_SZ * CV * NM;
    float* t0 = (float*)((char*)d_ws + base_bytes);
    float* t1 = t0 + t_elems;

    norms_kernel<<<(B_SZ * NM + 255) / 256, 256, 0, stream>>>(mk, norms);

    dim3 grid(NQ / QT, B_SZ);
    topk_kernel<<<grid, NWAVE * 32, 0, stream>>>(mk, qk, norms, w_ws, i_ws);

    if (ws_size >= base_bytes + 2u * t_elems * 4u) {
        dim3 tg(NM / 32, CV / 32, B_SZ);
        transpose_kernel<<<tg, 256, 0, stream>>>(mv0, t0);
        transpose_kernel<<<tg, 256, 0, stream>>>(mv1, t1);
        readout_t_kernel<<<B_SZ * NQ, CV, 0, stream>>>(t0, t1, w_ws, i_ws, out);
    } else {
        readout_kernel<<<B_SZ * NQ, CV, 0, stream>>>(mv0, mv1, w_ws, i_ws, out);
    }
}